// BlockAttention_53429393162349
// MI455X (gfx1250) — compile-verified
//
#include <hip/hip_runtime.h>
#include <math.h>

typedef __attribute__((ext_vector_type(2))) float v2f;
typedef __attribute__((ext_vector_type(8))) float v8f;

#define BM 128
#define BN 64
#define BK 32
#define LDA_S 36    // padded LDS stride (dwords) for A tile
#define LDBP 160    // dword stride per k-PAIR row of B tile (64*2 data + 32 pad
                    // -> second lane-half lands 32 banks away: conflict-free b64)
#define NTHREADS 256

__device__ __forceinline__ v8f wmma_f32(v2f a, v2f b, v8f c) {
  // D = A(16x4) * B(4x16) + C(16x16), fp32 matrix core
  return __builtin_amdgcn_wmma_f32_16x16x4_f32(
      /*neg_a=*/false, a, /*neg_b=*/false, b,
      /*c_mod=*/(short)0, c, /*reuse_a=*/false, /*reuse_b=*/false);
}

// Generic LDS-tiled fp32 WMMA GEMM: C[tile] = A(MxK,row-major) * B(KxN) [*scale][+bias]
// TRANS_B: B supplied as (N x K) row-major, loaded transposed into LDS (for Q*K^T).
// B tile LDS layout is k-pair interleaved: BsP[k>>1][n][k&1] so each WMMA B
// fragment (rows k,k+1 at col n) is one contiguous 8-byte ds_load_b64.
// Double-buffered: global loads for tile t+1 are issued before computing tile t,
// so HBM/L2 latency overlaps the 32 WMMAs of the current tile.
template <bool TRANS_B, bool HAS_BIAS>
__device__ void gemm_core(const float* __restrict__ A, int lda,
                          const float* __restrict__ B, int ldb,
                          float* __restrict__ C, int ldc, int K,
                          const float* __restrict__ bias, float scale,
                          int tileM, int tileN) {
  __shared__ float As[2][BM * LDA_S];
  __shared__ float Bs[2][(BK / 2) * LDBP];

  const int tid  = threadIdx.x;
  const int lane = tid & 31;
  const int wave = tid >> 5;      // 8 waves
  const int wm   = wave >> 1;     // 0..3  (M sub-tile)
  const int wn   = wave & 1;      // 0..1  (N sub-tile)
  const int row  = lane & 15;     // lane row/col within 16
  const int kh   = lane >> 4;     // lane half (K-split / M-split per ISA layout)

  v8f acc[2][2];
#pragma unroll
  for (int i = 0; i < 2; ++i)
#pragma unroll
    for (int j = 0; j < 2; ++j)
#pragma unroll
      for (int r = 0; r < 8; ++r) acc[i][j][r] = 0.0f;

  float4 ra[4];   // staged A tile chunk (BM x BK = 1024 f4 / 256 thr)
  float4 rb[2];   // staged B tile chunk (BK x BN =  512 f4 / 256 thr)

  auto g_load = [&](int k0) {
#pragma unroll
    for (int i = 0; i < 4; ++i) {
      int f4 = tid + i * NTHREADS;
      ra[i] = *(const float4*)(A + (size_t)(tileM + (f4 >> 3)) * lda + k0 + ((f4 & 7) << 2));
    }
    if (TRANS_B) {
#pragma unroll
      for (int i = 0; i < 2; ++i) {
        int f4 = tid + i * NTHREADS;
        rb[i] = *(const float4*)(B + (size_t)(tileN + (f4 >> 3)) * ldb + k0 + ((f4 & 7) << 2));
      }
    } else {
#pragma unroll
      for (int i = 0; i < 2; ++i) {
        int f4 = tid + i * NTHREADS;
        rb[i] = *(const float4*)(B + (size_t)(k0 + (f4 >> 4)) * ldb + tileN + ((f4 & 15) << 2));
      }
    }
  };

  auto l_store = [&](int buf) {
#pragma unroll
    for (int i = 0; i < 4; ++i) {
      int f4 = tid + i * NTHREADS;
      float* dst = &As[buf][(f4 >> 3) * LDA_S + ((f4 & 7) << 2)];
      dst[0] = ra[i].x; dst[1] = ra[i].y; dst[2] = ra[i].z; dst[3] = ra[i].w;
    }
    if (TRANS_B) {
#pragma unroll
      for (int i = 0; i < 2; ++i) {
        int f4 = tid + i * NTHREADS;
        int n = f4 >> 3, kq = (f4 & 7) << 2;
        float* d0 = &Bs[buf][(kq >> 1) * LDBP + n * 2];
        d0[0] = rb[i].x; d0[1] = rb[i].y;           // rows kq, kq+1
        d0[LDBP] = rb[i].z; d0[LDBP + 1] = rb[i].w; // rows kq+2, kq+3
      }
    } else {
#pragma unroll
      for (int i = 0; i < 2; ++i) {
        int f4 = tid + i * NTHREADS;
        int r = f4 >> 4, c = (f4 & 15) << 2;
        float* d0 = &Bs[buf][(r >> 1) * LDBP + c * 2 + (r & 1)];
        d0[0] = rb[i].x; d0[2] = rb[i].y; d0[4] = rb[i].z; d0[6] = rb[i].w;
      }
    }
  };

  // prologue: stage tile 0
  g_load(0);
  l_store(0);
  __syncthreads();

  const int ntiles = K / BK;
  for (int t = 0; t < ntiles; ++t) {
    const int cur  = t & 1;
    const bool more = (t + 1) < ntiles;
    if (more) {
      g_load((t + 1) * BK);                       // overlaps with WMMAs below
      if (t + 2 < ntiles)                          // pull tile t+2 toward L2
        __builtin_prefetch(A + (size_t)(tileM + (tid >> 3)) * lda + (t + 2) * BK, 0, 1);
    }

    const float* Ab = As[cur];
    const float* Bb = Bs[cur];
#pragma unroll
    for (int kk = 0; kk < BK; kk += 4) {
      const int kq = kk + 2 * kh;  // lanes 0-15: K=kk,kk+1 ; lanes 16-31: K=kk+2,kk+3
      const float* a0p = &Ab[(wm * 32 + row) * LDA_S + kq];
      const float* a1p = a0p + 16 * LDA_S;
      v2f a0, a1, b0, b1;
      a0.x = a0p[0]; a0.y = a0p[1];
      a1.x = a1p[0]; a1.y = a1p[1];
      // contiguous b64 fragments: (row kq, row kq+1) at col, col+16
      const float* bp = &Bb[(kq >> 1) * LDBP + (wn * 32 + row) * 2];
      b0.x = bp[0];  b0.y = bp[1];
      b1.x = bp[32]; b1.y = bp[33];
      acc[0][0] = wmma_f32(a0, b0, acc[0][0]);
      acc[0][1] = wmma_f32(a0, b1, acc[0][1]);
      acc[1][0] = wmma_f32(a1, b0, acc[1][0]);
      acc[1][1] = wmma_f32(a1, b1, acc[1][1]);
    }

    if (more) {
      __syncthreads();          // all waves done reading buf[cur^1]'s last use
      l_store(cur ^ 1);         // stage tile t+1
      __syncthreads();          // tile t+1 visible before next iteration reads it
    }
  }

  // ---- epilogue: C/D layout -> VGPR r holds M = r + 8*kh, N = lane&15 ----
  const int col = lane & 15;
#pragma unroll
  for (int i = 0; i < 2; ++i) {
#pragma unroll
    for (int j = 0; j < 2; ++j) {
      const int r0 = tileM + wm * 32 + i * 16 + 8 * kh;
      const int cc = tileN + wn * 32 + j * 16 + col;
      const float badd = HAS_BIAS ? bias[cc] : 0.0f;
#pragma unroll
      for (int r = 0; r < 8; ++r)
        C[(size_t)(r0 + r) * ldc + cc] = acc[i][j][r] * scale + badd;
    }
  }
}

// ---- kernels ----

__global__ __launch_bounds__(NTHREADS)
void k_gemm_bias(const float* __restrict__ A, const float* __restrict__ W,
                 const float* __restrict__ bias, float* __restrict__ C,
                 int K, int N) {
  gemm_core<false, true>(A, K, W, N, C, N, K, bias, 1.0f,
                         blockIdx.y * BM, blockIdx.x * BN);
}

__global__ __launch_bounds__(NTHREADS)
void k_qkt(const float* __restrict__ Q, const float* __restrict__ Km,
           float* __restrict__ S, float scale) {
  const int g = blockIdx.z;               // 128 groups: ((b*16+blk)*2+which)
  const int b = g >> 5, blk = (g >> 1) & 15, which = g & 1;
  const int kblk = which ? (blk ? blk - 1 : 0) : blk;   // cross -> prev block (0 -> itself)
  const float* Ap = Q  + (size_t)(b * 4096 + blk  * 256) * 1024;
  const float* Bp = Km + (size_t)(b * 4096 + kblk * 256) * 1024;
  float* Cp = S + (size_t)g * 256 * 256;
  gemm_core<true, false>(Ap, 1024, Bp, 1024, Cp, 256, 1024, nullptr, scale,
                         blockIdx.y * BM, blockIdx.x * BN);
}

__global__ __launch_bounds__(NTHREADS)
void k_softmax256(float* __restrict__ S) {
  const int lane = threadIdx.x & 31;
  const int wave = threadIdx.x >> 5;      // one wave32 per 256-wide row
  float* r = S + ((size_t)blockIdx.x * 8 + wave) * 256;
  float v[8];
  float mx = -3.402823466e38f;
#pragma unroll
  for (int i = 0; i < 8; ++i) { v[i] = r[lane + 32 * i]; mx = fmaxf(mx, v[i]); }
#pragma unroll
  for (int off = 16; off >= 1; off >>= 1) mx = fmaxf(mx, __shfl_xor(mx, off, 32));
  float sum = 0.0f;
#pragma unroll
  for (int i = 0; i < 8; ++i) { v[i] = expf(v[i] - mx); sum += v[i]; }
#pragma unroll
  for (int off = 16; off >= 1; off >>= 1) sum += __shfl_xor(sum, off, 32);
  const float inv = 1.0f / sum;
#pragma unroll
  for (int i = 0; i < 8; ++i) r[lane + 32 * i] = v[i] * inv;
}

__global__ __launch_bounds__(NTHREADS)
void k_pv(const float* __restrict__ P, const float* __restrict__ V,
          float* __restrict__ Cat) {
  const int g = blockIdx.z;
  const int b = g >> 5, blk = (g >> 1) & 15, which = g & 1;
  const int kblk = which ? (blk ? blk - 1 : 0) : blk;
  const float* Ap = P + (size_t)g * 256 * 256;
  const float* Bp = V + (size_t)(b * 4096 + kblk * 256) * 1024;
  float* Cp = Cat + (size_t)((b * 16 + blk) * 512 + which * 256) * 1024;
  gemm_core<false, false>(Ap, 256, Bp, 1024, Cp, 1024, 256, nullptr, 1.0f,
                          blockIdx.y * BM, blockIdx.x * BN);
}

extern "C" void kernel_launch(void* const* d_in, const int* in_sizes, int n_in,
                              void* d_out, int out_size, void* d_ws, size_t ws_size,
                              hipStream_t stream) {
  (void)in_sizes; (void)n_in; (void)out_size; (void)ws_size;
  const float* x  = (const float*)d_in[0];
  const float* wq = (const float*)d_in[1];
  const float* bq = (const float*)d_in[2];
  const float* wk = (const float*)d_in[3];
  const float* bk = (const float*)d_in[4];
  const float* wv = (const float*)d_in[5];
  const float* bv = (const float*)d_in[6];
  const float* wo = (const float*)d_in[7];
  const float* bo = (const float*)d_in[8];
  float* out = (float*)d_out;

  const int D = 1024, Bsz = 4, N = 4096;
  const size_t QSZ = (size_t)Bsz * N * D;     // 16.78M floats each
  float* Q   = (float*)d_ws;
  float* Km  = Q + QSZ;
  float* V   = Km + QSZ;
  float* S   = V + QSZ;                        // 128 * 256*256 floats
  float* Cat = Q;                              // aliases Q||K (dead after k_qkt)

  const float scale = 0.03125f;               // 1/sqrt(1024)
  dim3 blk(NTHREADS);

  dim3 gproj(D / BN, (Bsz * N) / BM);         // (16, 128)
  k_gemm_bias<<<gproj, blk, 0, stream>>>(x, wq, bq, Q,  D, D);
  k_gemm_bias<<<gproj, blk, 0, stream>>>(x, wk, bk, Km, D, D);
  k_gemm_bias<<<gproj, blk, 0, stream>>>(x, wv, bv, V,  D, D);

  dim3 gqkt(256 / BN, 256 / BM, 128);         // (4, 2, 128)
  k_qkt<<<gqkt, blk, 0, stream>>>(Q, Km, S, scale);

  k_softmax256<<<dim3(32768 / 8), blk, 0, stream>>>(S);

  dim3 gpv(D / BN, 256 / BM, 128);            // (16, 2, 128)
  k_pv<<<gpv, blk, 0, stream>>>(S, V, Cat);

  dim3 gout(D / BN, 32768 / BM);              // (16, 256)
  k_gemm_bias<<<gout, blk, 0, stream>>>(Cat, wo, bo, out, D, D);
}